// STTN_1778116460927
// MI455X (gfx1250) — compile-verified
//
#include <hip/hip_runtime.h>

// ---- problem constants ------------------------------------------------
#define BB 4
#define NN 1024
#define TT 12
#define EE 64
#define HH 4
#define HD 16
#define N2 2048
#define BT (BB*TT)          // 48
#define BTH (BB*TT*HH)      // 192
#define MAXDEG 64
#define CSTF 1e-05f

typedef __attribute__((ext_vector_type(2))) float v2f;
typedef __attribute__((ext_vector_type(8))) float v8f;
typedef int ai4 __attribute__((vector_size(4 * sizeof(int))));   // b128 payload type
typedef __attribute__((address_space(1))) ai4* gas_ai4;
typedef __attribute__((address_space(3))) ai4* las_ai4;

#if __has_builtin(__builtin_amdgcn_global_load_async_to_lds_b128)
#define HAVE_ASYNC_LDS 1
#else
#define HAVE_ASYNC_LDS 0
#warning "gfx1250 async global->LDS builtin not available; using synchronous LDS staging"
#endif

__device__ inline void async_copy16(const float* g, float* l) {
#if HAVE_ASYNC_LDS
    __builtin_amdgcn_global_load_async_to_lds_b128(
        (gas_ai4)(ai4*)g, (las_ai4)(ai4*)l, 0, 0);
#else
    *(float4*)l = *(const float4*)g;
#endif
}

__device__ inline void async_wait_all() {
#if HAVE_ASYNC_LDS
#if __has_builtin(__builtin_amdgcn_s_wait_asynccnt)
    __builtin_amdgcn_s_wait_asynccnt(0);
#else
    asm volatile("s_wait_asynccnt 0x0" ::: "memory");
#endif
#endif
}

// 16-lane (half-wave) sum reduction, wave32
__device__ inline float dot16(float v) {
    v += __shfl_xor(v, 1, 16);
    v += __shfl_xor(v, 2, 16);
    v += __shfl_xor(v, 4, 16);
    v += __shfl_xor(v, 8, 16);
    return v;
}

// Y(16x16) = A(16x16) * W^T, A rows strided by arowstride, W row-major [16][16].
// B[k][n] = W[n][k]. Uses V_WMMA_F32_16X16X4_F32 chained over K=16.
__device__ inline v8f gemm16(const float* Abase, size_t arowstride,
                             const float* W, int row, int khalf) {
    v8f acc = {};
    #pragma unroll
    for (int j = 0; j < 4; ++j) {
        int kk = 4*j + 2*khalf;
        v2f a = *(const v2f*)(Abase + (size_t)row*arowstride + kk);
        v2f b = *(const v2f*)(W + row*16 + kk);
        acc = __builtin_amdgcn_wmma_f32_16x16x4_f32(false, a, false, b,
                                                    (short)0, acc, false, false);
    }
    return acc;
}

// ---- graph preprocessing: CSC row lists + inverse degree ---------------
__global__ void build_graph_kernel(const float* __restrict__ adj,
                                   int* __restrict__ rows, int* __restrict__ cnt,
                                   float* __restrict__ dinv) {
    int c = blockIdx.x * blockDim.x + threadIdx.x;
    if (c >= NN) return;
    int k = 0, tot = 0;
    for (int r = 0; r < NN; ++r) {
        if (adj[(size_t)r*NN + c] != 0.f) {
            ++tot;
            if (k < MAXDEG) rows[c*MAXDEG + (k++)] = r;
        }
    }
    cnt[c] = k;
    dinv[c] = 1.f / (float)(tot + 1);   // +1 identity block edge
}

// ---- stage 1 projections (v,k,q) + hidden init -------------------------
__global__ void proj1_kernel(const float* __restrict__ values,
                             const float* __restrict__ keys,
                             const float* __restrict__ query,
                             const float* __restrict__ Wv, const float* __restrict__ Wk,
                             const float* __restrict__ Wq,
                             const float* __restrict__ hopwise,
                             float* __restrict__ vh, float* __restrict__ kh,
                             float* __restrict__ qh, float* __restrict__ hidden) {
    int lane = threadIdx.x & 31, wv = threadIdx.x >> 5;
    int gw = blockIdx.x * 8 + wv;               // BTH*64 waves
    int tile = gw & 63, bth = gw >> 6;
    int h = bth & 3, bt = bth >> 2, t = bt % TT, b = bt / TT;
    int n0 = tile * 16, row = lane & 15, khalf = lane >> 4;
    size_t ibase = ((size_t)(b*NN + n0)*TT + t)*EE + h*HD;
    size_t obase = (size_t)bth*NN*HD + (size_t)n0*HD;
    float hop0 = hopwise[0];
    v8f av = gemm16(values + ibase, (size_t)TT*EE, Wv, row, khalf);
    v8f ak = gemm16(keys   + ibase, (size_t)TT*EE, Wk, row, khalf);
    v8f aq = gemm16(query  + ibase, (size_t)TT*EE, Wq, row, khalf);
    #pragma unroll
    for (int r = 0; r < 8; ++r) {
        int m = r + 8*khalf;
        size_t o = obase + (size_t)m*HD + row;
        float v = av[r];
        vh[o] = v;
        hidden[o] = v * hop0;
        kh[o] = fmaxf(ak[r], 0.f);
        qh[o] = fmaxf(aq[r], 0.f);
    }
}

// ---- teleport: teleM = k^T v / N (WMMA over nodes), teleK = sum(k)/N ---
__global__ void telem_kernel(const float* __restrict__ kh, const float* __restrict__ vh,
                             float* __restrict__ teleM, float* __restrict__ teleK) {
    __shared__ float redM[8][256];
    __shared__ float redK[256];
    int bth = blockIdx.x;
    int lane = threadIdx.x & 31, wv = threadIdx.x >> 5;
    int row = lane & 15, khalf = lane >> 4;
    size_t base = (size_t)bth*NN*HD;
    v8f acc = {};
    for (int step = 0; step < 32; ++step) {
        int n0 = 4 * (wv + 8*step);
        size_t ia = base + (size_t)(n0 + 2*khalf)*HD + row;
        v2f a = { kh[ia], kh[ia + HD] };   // A[d][node]
        v2f b = { vh[ia], vh[ia + HD] };   // B[node][z]
        acc = __builtin_amdgcn_wmma_f32_16x16x4_f32(false, a, false, b,
                                                    (short)0, acc, false, false);
    }
    #pragma unroll
    for (int r = 0; r < 8; ++r) redM[wv][(r + 8*khalf)*16 + row] = acc[r];
    {   // teleK partials: thread = (chunk of 64 nodes, dim d)
        int d = threadIdx.x & 15, chunk = threadIdx.x >> 4;
        float s = 0.f;
        for (int i = 0; i < 64; ++i) s += kh[base + (size_t)(chunk*64 + i)*HD + d];
        redK[threadIdx.x] = s;
    }
    __syncthreads();
    float s = 0.f;
    #pragma unroll
    for (int w = 0; w < 8; ++w) s += redM[w][threadIdx.x];
    teleM[(size_t)bth*256 + threadIdx.x] = s / (float)NN;
    if (threadIdx.x < 16) {
        float sk = 0.f;
        for (int ch = 0; ch < 16; ++ch) sk += redK[ch*16 + threadIdx.x];
        teleK[bth*16 + threadIdx.x] = sk / (float)NN;
    }
}

// ---- teleport apply: hidden += (q @ teleM) / (q . teleK + CST) * tp ----
__global__ void tele_apply_kernel(const float* __restrict__ qh,
                                  const float* __restrict__ teleM,
                                  const float* __restrict__ teleK,
                                  const float* __restrict__ teleport,
                                  float* __restrict__ hidden) {
    __shared__ float Cs[8][16];
    int lane = threadIdx.x & 31, wv = threadIdx.x >> 5;
    int gw = blockIdx.x * 8 + wv;               // BTH*64 waves
    int tile = gw & 63, bth = gw >> 6;
    int n0 = tile * 16, row = lane & 15, khalf = lane >> 4;
    size_t base = (size_t)bth*NN*HD;
    const float* tM = teleM + (size_t)bth*256;
    v8f acc = {};
    #pragma unroll
    for (int j = 0; j < 4; ++j) {
        int kk = 4*j + 2*khalf;
        v2f a = *(const v2f*)(qh + base + (size_t)(n0 + row)*HD + kk);
        v2f b = { tM[kk*16 + row], tM[(kk+1)*16 + row] };
        acc = __builtin_amdgcn_wmma_f32_16x16x4_f32(false, a, false, b,
                                                    (short)0, acc, false, false);
    }
    if (lane < 16) {
        const float* tK = teleK + bth*16;
        float s = CSTF;
        #pragma unroll
        for (int d = 0; d < 16; ++d) s += qh[base + (size_t)(n0 + lane)*HD + d] * tK[d];
        Cs[wv][lane] = s;
    }
    __syncthreads();
    float tp = teleport[0];
    #pragma unroll
    for (int r = 0; r < 8; ++r) {
        int m = r + 8*khalf;
        hidden[base + (size_t)(n0 + m)*HD + row] += acc[r] / Cs[wv][m] * tp;
    }
}

// ---- stage 2 projections over 2N nodes (prev,cur timesteps) ------------
__global__ void proj2_kernel(const float* __restrict__ values,
                             const float* __restrict__ keys,
                             const float* __restrict__ query,
                             const float* __restrict__ Wv2, const float* __restrict__ Wk2,
                             const float* __restrict__ Wq2,
                             float* __restrict__ v2h, float* __restrict__ k2h,
                             float* __restrict__ q2h) {
    int lane = threadIdx.x & 31, wv = threadIdx.x >> 5;
    int gw = blockIdx.x * 8 + wv;               // BTH*128 waves
    int tile = gw & 127, bth = gw >> 7;
    int h = bth & 3, bt = bth >> 2, t = bt % TT, b = bt / TT;
    int c0 = tile * 16, row = lane & 15, khalf = lane >> 4;
    int tsrc  = (c0 < NN) ? ((t > 0) ? t - 1 : 0) : t;   // prev half / cur half
    int nbase = (c0 < NN) ? c0 : (c0 - NN);
    size_t ibase = ((size_t)(b*NN + nbase)*TT + tsrc)*EE + h*HD;
    size_t obase = (size_t)bth*N2*HD + (size_t)c0*HD;
    v8f av = gemm16(values + ibase, (size_t)TT*EE, Wv2, row, khalf);
    v8f ak = gemm16(keys   + ibase, (size_t)TT*EE, Wk2, row, khalf);
    v8f aq = gemm16(query  + ibase, (size_t)TT*EE, Wq2, row, khalf);
    #pragma unroll
    for (int r = 0; r < 8; ++r) {
        int m = r + 8*khalf;
        size_t o = obase + (size_t)m*HD + row;
        v2h[o] = av[r];
        k2h[o] = fmaxf(ak[r], 0.f);
        q2h[o] = fmaxf(aq[r], 0.f);
    }
}

// ---- propagate k: dst[c] = sum_r P[c,r] src[r] (block adjacency) -------
__global__ void prop_k_kernel(const float* __restrict__ src, float* __restrict__ dst,
                              const int* __restrict__ rows, const int* __restrict__ cnt,
                              const float* __restrict__ dinv) {
    size_t idx = (size_t)blockIdx.x * 256 + threadIdx.x;   // BTH*N2*16 threads
    int d = idx & 15;
    int c = (idx >> 4) & (N2 - 1);
    int bth = idx >> 15;
    int cp = c & (NN - 1);
    bool lower = (c < NN);
    const float* s = src + (size_t)bth*N2*HD;
    // identity-block edge (source is the node's mirror in the other half)
    float acc = dinv[cp] * s[(size_t)(lower ? NN + cp : cp)*HD + d];
    int m = cnt[cp];
    const int* rl = rows + cp*MAXDEG;
    int off = lower ? 0 : NN;
    for (int i = 0; i < m; ++i) {
        int r = rl[i];
        acc += dinv[r] * s[(size_t)(off + r)*HD + d];
    }
    dst[(size_t)bth*N2*HD + (size_t)c*HD + d] = acc;
}

// edge accumulate against LDS-resident k/v slabs
__device__ inline void edge_accum_lds(float qd, const float* ksh, const float* vsh,
                                      int r, float w, int lz, float& acc) {
    float s = dot16(qd * ksh[r*HD + lz]);
    acc += w * s * vsh[r*HD + lz];
}

// ---- fused 2-hop propagation, k/v slabs staged in LDS ------------------
// One workgroup per (b,t,h): stage k0/v0 (2 x 128KB) + dinv/cnt into LDS via
// async global->LDS DMA, then compute both hop contributions from LDS.
//   Hh1[c] = sum_r P[c,r] (q[c].k0[r]) v0[r]
//   Hh2[c] = sum_m P[c,m] sum_r P[m,r] (q[c].k0[r]) v0[r]
__global__ void __launch_bounds__(1024)
hop_fused_kernel(const float* __restrict__ q2h, const float* __restrict__ k2h,
                 const float* __restrict__ v2h,
                 const float* __restrict__ k1, const float* __restrict__ k2,
                 const int* __restrict__ rows, const int* __restrict__ cnt,
                 const float* __restrict__ dinv,
                 const float* __restrict__ hopwise,
                 float* __restrict__ hidden) {
    extern __shared__ float smem[];
    float* ksh = smem;                         // N2*HD = 32768 floats (128KB)
    float* vsh = smem + N2*HD;                 // 128KB
    float* dsh = smem + 2*N2*HD;               // NN floats
    int*   csh = (int*)(smem + 2*N2*HD + NN);  // NN ints
    int bth = blockIdx.x;
    size_t base = (size_t)bth*N2*HD;
    {
        const float* gk = k2h + base;
        const float* gv = v2h + base;
        for (int i = threadIdx.x; i < (N2*HD)/4; i += 1024) {
            async_copy16(gk + 4*i, ksh + 4*i);
            async_copy16(gv + 4*i, vsh + 4*i);
        }
        dsh[threadIdx.x] = dinv[threadIdx.x];   // blockDim == NN == 1024
        csh[threadIdx.x] = cnt[threadIdx.x];
        async_wait_all();
        __syncthreads();
    }
    int g = threadIdx.x >> 4, lz = threadIdx.x & 15;   // 64 lane-groups of 16
    float hw1 = hopwise[1], hw2 = hopwise[2];
    for (int it = 0; it < 16; ++it) {
        int cp = it*64 + g;
        int c2 = NN + cp;
        float qd = q2h[base + (size_t)c2*HD + lz];
        int mc = csh[cp];
        const int* rlc = rows + cp*MAXDEG;
        float dc = dsh[cp];
        // ---- hop 1 ----
        float acc1 = 0.f;
        edge_accum_lds(qd, ksh, vsh, cp, dc, lz, acc1);            // identity edge
        for (int i = 0; i < mc; ++i) {
            int rp = rlc[i];
            edge_accum_lds(qd, ksh, vsh, NN + rp, dsh[rp], lz, acc1);
        }
        float C1 = dot16(qd * k1[base + (size_t)c2*HD + lz]) + CSTF;
        // ---- hop 2 ----
        float acc2 = 0.f;
        {   // middle node m = cp (lower half), w1 = dinv[cp]
            edge_accum_lds(qd, ksh, vsh, NN + cp, dc*dc, lz, acc2);     // r = N+cp
            for (int i = 0; i < mc; ++i) {
                int r = rlc[i];
                edge_accum_lds(qd, ksh, vsh, r, dc*dsh[r], lz, acc2);   // r lower half
            }
        }
        for (int i = 0; i < mc; ++i) {          // middle nodes m = N+m'
            int mp = rlc[i];
            float w1 = dsh[mp];
            edge_accum_lds(qd, ksh, vsh, mp, w1*w1, lz, acc2);          // r = m'
            int mm = csh[mp];
            const int* rlm = rows + mp*MAXDEG;
            for (int j = 0; j < mm; ++j) {
                int rp = rlm[j];
                edge_accum_lds(qd, ksh, vsh, NN + rp, w1*dsh[rp], lz, acc2);
            }
        }
        float C2 = dot16(qd * k2[base + (size_t)c2*HD + lz]) + CSTF;
        hidden[(size_t)bth*NN*HD + (size_t)cp*HD + lz] += acc1/C1*hw1 + acc2/C2*hw2;
    }
}

// ---- output projection: out[b,n,t,:] = hidden[b,t,n,:] @ Wout^T + bout -
__global__ void outproj_kernel(const float* __restrict__ hidden,
                               const float* __restrict__ Wout,
                               const float* __restrict__ bout,
                               float* __restrict__ out) {
    int lane = threadIdx.x & 31, wv = threadIdx.x >> 5;
    int gw = blockIdx.x * 8 + wv;               // BT*64*4 waves
    int eblk = gw & 3, tile = (gw >> 2) & 63, bt = gw >> 8;
    int t = bt % TT, b = bt / TT;
    int n0 = tile*16, e0 = eblk*16, row = lane & 15, khalf = lane >> 4;
    v8f acc = {};
    #pragma unroll
    for (int j = 0; j < 16; ++j) {              // K = 64 input channels
        int kk = 4*j + 2*khalf;
        int h = kk >> 4, dd = kk & 15;
        v2f a = *(const v2f*)(hidden + (((size_t)(bt*HH + h)*NN + n0 + row)*HD + dd));
        v2f bm = *(const v2f*)(Wout + (size_t)(e0 + row)*EE + kk);
        acc = __builtin_amdgcn_wmma_f32_16x16x4_f32(false, a, false, bm,
                                                    (short)0, acc, false, false);
    }
    #pragma unroll
    for (int r = 0; r < 8; ++r) {
        int m = r + 8*khalf;
        int n = n0 + m, e = e0 + row;
        out[(((size_t)b*NN + n)*TT + t)*EE + e] = acc[r] + bout[e];
    }
}

extern "C" void kernel_launch(void* const* d_in, const int* in_sizes, int n_in,
                              void* d_out, int out_size, void* d_ws, size_t ws_size,
                              hipStream_t stream) {
    const float* values  = (const float*)d_in[0];
    const float* keys    = (const float*)d_in[1];
    const float* query   = (const float*)d_in[2];
    const float* adj     = (const float*)d_in[3];
    const float* Wv      = (const float*)d_in[4];
    const float* Wk      = (const float*)d_in[5];
    const float* Wq      = (const float*)d_in[6];
    const float* Wv2     = (const float*)d_in[7];
    const float* Wk2     = (const float*)d_in[8];
    const float* Wq2     = (const float*)d_in[9];
    const float* Wout    = (const float*)d_in[10];
    const float* bout    = (const float*)d_in[11];
    const float* hopwise = (const float*)d_in[12];
    const float* teleport= (const float*)d_in[13];
    float* out = (float*)d_out;

    // workspace carve-up (~177 MB of f32 + 260 KB of int)
    const size_t SZ1 = (size_t)BTH*NN*HD;   // 3,145,728
    const size_t SZ2 = (size_t)BTH*N2*HD;   // 6,291,456
    float* w = (float*)d_ws;
    float* hidden = w;              w += SZ1;
    float* vh     = w;              w += SZ1;
    float* kh     = w;              w += SZ1;
    float* qh     = w;              w += SZ1;
    float* teleM  = w;              w += (size_t)BTH*256;
    float* teleK  = w;              w += (size_t)BTH*16;
    float* v2h    = w;              w += SZ2;
    float* k2h    = w;              w += SZ2;
    float* q2h    = w;              w += SZ2;
    float* k1b    = w;              w += SZ2;
    float* k2b    = w;              w += SZ2;
    float* dinv   = w;              w += NN;
    int*   rowsl  = (int*)w;
    int*   cntl   = rowsl + NN*MAXDEG;

    // 2x128KB k/v slabs + dinv + cnt in LDS (270336 B <= 320KB WGP LDS)
    const size_t lds_bytes = (size_t)(2*N2*HD + NN)*sizeof(float) + NN*sizeof(int);
    (void)hipFuncSetAttribute((const void*)hop_fused_kernel,
                              hipFuncAttributeMaxDynamicSharedMemorySize,
                              (int)lds_bytes);

    build_graph_kernel<<<NN/256, 256, 0, stream>>>(adj, rowsl, cntl, dinv);
    proj1_kernel<<<BTH*64/8, 256, 0, stream>>>(values, keys, query, Wv, Wk, Wq,
                                               hopwise, vh, kh, qh, hidden);
    telem_kernel<<<BTH, 256, 0, stream>>>(kh, vh, teleM, teleK);
    tele_apply_kernel<<<BTH*64/8, 256, 0, stream>>>(qh, teleM, teleK, teleport, hidden);
    proj2_kernel<<<BTH*128/8, 256, 0, stream>>>(values, keys, query, Wv2, Wk2, Wq2,
                                                v2h, k2h, q2h);
    prop_k_kernel<<<(unsigned)(SZ2/256), 256, 0, stream>>>(k2h, k1b, rowsl, cntl, dinv);
    prop_k_kernel<<<(unsigned)(SZ2/256), 256, 0, stream>>>(k1b, k2b, rowsl, cntl, dinv);
    hop_fused_kernel<<<BTH, 1024, lds_bytes, stream>>>(q2h, k2h, v2h, k1b, k2b,
                                                       rowsl, cntl, dinv, hopwise, hidden);
    outproj_kernel<<<BT*256/8, 256, 0, stream>>>(hidden, Wout, bout, out);
}